// PowerGridModel_47820165874391
// MI455X (gfx1250) — compile-verified
//
#include <hip/hip_runtime.h>
#include <hip/hip_bf16.h>

// ---------------------------------------------------------------------------
// Graph Transformer (PyG TransformerConv x3 + mean pool + FC) for gfx1250.
//
// Design notes (MI455X):
//  * Dense GEMMs use V_WMMA_F32_16X16X4_F32 (full fp32 precision; FLOPs are
//    tiny so the fp32 WMMA rate is more than enough).
//  * GEMM A-loads are UNGUARDED with a clamped row address: OOB rows only
//    corrupt C rows that are masked at the store, so the inner loop is pure
//    load+wmma with no EXEC manipulation.
//  * Edge embedding e = edge_attr @ We is NEVER materialized (saves ~1 GB of
//    traffic per layer); We (26x256 = 26.6KB) lives in LDS and e is recomputed
//    per edge with 26 FMAs per channel.
//  * Segment softmax via ordered-uint atomicMax, then unnormalized
//    exp(l-m)*v_j accumulation with float atomicAdd, normalized per node.
//    Q/K/V/OutAcc working set (~82MB) is L2-resident (192MB L2).
// ---------------------------------------------------------------------------

#define N_NODES   20000
#define N_EDGES   320000
#define F_IN      128
#define EDGE_DIM  26
#define N_HEADS   8
#define HEAD_D    32
#define HD        256
#define N_GRAPHS  64
#define N_CLASSES 32

typedef float v2f __attribute__((ext_vector_type(2)));
typedef float v8f __attribute__((ext_vector_type(8)));

// ---------------- helpers ----------------

__device__ __forceinline__ unsigned fkey(float f) {
    unsigned u = __float_as_uint(f);
    return (u & 0x80000000u) ? ~u : (u | 0x80000000u);
}
// inverse of fkey: MSB set -> original non-negative; else original negative
__device__ __forceinline__ float funkey2(unsigned u) {
    return (u & 0x80000000u) ? __uint_as_float(u & 0x7fffffffu)
                             : __uint_as_float(~u);
}
// key 0 is below every real float's key, so zero-init is the atomicMax identity.

__global__ void zero_f32_kernel(float* p, long long n) {
    long long i = (long long)blockIdx.x * blockDim.x + threadIdx.x;
    long long stride = (long long)gridDim.x * blockDim.x;
    for (; i < n; i += stride) p[i] = 0.0f;
}

// ---------------- fused QKVS GEMM: O = X @ W + b, WMMA fp32 ----------------
// grid: (ceil(nrows/64), 16, 4)   block: 128 (4 waves, 1 tile each)
__global__ __launch_bounds__(128)
void gemm4_kernel(const float* __restrict__ X, int fin, int nrows,
                  const float* W0, const float* B0, float* O0,
                  const float* W1, const float* B1, float* O1,
                  const float* W2, const float* B2, float* O2,
                  const float* W3, const float* B3, float* O3) {
    const float* W; const float* Bv; float* O;
    switch (blockIdx.z) {
        case 0:  W = W0; Bv = B0; O = O0; break;
        case 1:  W = W1; Bv = B1; O = O1; break;
        case 2:  W = W2; Bv = B2; O = O2; break;
        default: W = W3; Bv = B3; O = O3; break;
    }
    const int lane = threadIdx.x & 31;
    const int wave = threadIdx.x >> 5;
    const int tm = (blockIdx.x * 4 + wave) * 16;   // tile row base
    const int tn = blockIdx.y * 16;                // tile col base
    const int r  = lane & 15;
    const int k0 = (lane >> 4) << 1;               // 0 or 2 (K split across half-waves)
    const int row = tm + r;
    const int col = tn + r;
    // Clamp OOB rows to row 0 and load unconditionally: the garbage only lands
    // in C rows that the store guard discards.  Keeps EXEC untouched in the loop.
    const int rowL = (row < nrows) ? row : 0;
    const float* xrow = X + (long long)rowL * fin + k0;
    const float* wcol = W + col + (long long)k0 * HD;

    v8f acc = {};
    for (int k = 0; k < fin; k += 4) {
        v2f a, b;
        float2 t = *(const float2*)(xrow + k);     // 8B aligned (k+k0 even)
        a.x = t.x; a.y = t.y;
        b.x = wcol[(long long)k * HD];
        b.y = wcol[(long long)k * HD + HD];
        acc = __builtin_amdgcn_wmma_f32_16x16x4_f32(
            false, a, false, b, (short)0, acc, false, false);
    }
    const float bias = Bv[col];
    const int rbase = tm + ((lane >> 4) << 3);     // lanes 16-31 hold M = i+8
    #pragma unroll
    for (int i = 0; i < 8; i++) {
        int rr = rbase + i;
        if (rr < nrows) O[(long long)rr * HD + col] = acc[i] + bias;
    }
}

// ---------------- edge pass 1: logits + segment max ----------------
#define EPW 8   // edges per wave
__global__ __launch_bounds__(256)
void edge_logits_kernel(const int* __restrict__ ei,
                        const float* __restrict__ ea,
                        const float* __restrict__ Q,
                        const float* __restrict__ K,
                        const float* __restrict__ We,
                        float* __restrict__ logits,
                        unsigned* __restrict__ mkey, int E) {
    __shared__ float sWe[EDGE_DIM * HD];
    for (int i = threadIdx.x; i < EDGE_DIM * HD; i += blockDim.x) sWe[i] = We[i];
    __syncthreads();

    const int lane = threadIdx.x & 31;
    const int wave = threadIdx.x >> 5;
    int ebase = (blockIdx.x * 8 + wave) * EPW;
    for (int t = 0; t < EPW; t++) {
        int e = ebase + t;
        if (e >= E) break;                       // uniform within wave
        int src = ei[e];
        int dst = ei[E + e];
        float ear = (lane < EDGE_DIM) ? ea[(long long)e * EDGE_DIM + lane] : 0.0f;
        float ev[N_HEADS];
        #pragma unroll
        for (int h = 0; h < N_HEADS; h++) ev[h] = 0.0f;
        for (int j = 0; j < EDGE_DIM; j++) {
            float aj = __shfl(ear, j, 32);
            #pragma unroll
            for (int h = 0; h < N_HEADS; h++)
                ev[h] = fmaf(aj, sWe[j * HD + h * 32 + lane], ev[h]);
        }
        const float* qr = Q + (long long)dst * HD;
        const float* kr = K + (long long)src * HD;
        #pragma unroll
        for (int h = 0; h < N_HEADS; h++) {
            float p = qr[h * 32 + lane] * (kr[h * 32 + lane] + ev[h]);
            p += __shfl_xor(p, 16); p += __shfl_xor(p, 8);
            p += __shfl_xor(p, 4);  p += __shfl_xor(p, 2);
            p += __shfl_xor(p, 1);
            if (lane == 0) {
                float lg = p * 0.17677669529663689f;   // 1/sqrt(32)
                logits[(long long)e * N_HEADS + h] = lg;
                atomicMax(&mkey[(long long)dst * N_HEADS + h], fkey(lg));
            }
        }
    }
}

// ---------------- edge pass 2: exp, denom, scatter of a * v_j ----------------
__global__ __launch_bounds__(256)
void edge_accum_kernel(const int* __restrict__ ei,
                       const float* __restrict__ ea,
                       const float* __restrict__ V,
                       const float* __restrict__ We,
                       const float* __restrict__ logits,
                       const unsigned* __restrict__ mkey,
                       float* __restrict__ denom,
                       float* __restrict__ outacc, int E) {
    __shared__ float sWe[EDGE_DIM * HD];
    for (int i = threadIdx.x; i < EDGE_DIM * HD; i += blockDim.x) sWe[i] = We[i];
    __syncthreads();

    const int lane = threadIdx.x & 31;
    const int wave = threadIdx.x >> 5;
    int ebase = (blockIdx.x * 8 + wave) * EPW;
    for (int t = 0; t < EPW; t++) {
        int e = ebase + t;
        if (e >= E) break;
        int src = ei[e];
        int dst = ei[E + e];
        float ear = (lane < EDGE_DIM) ? ea[(long long)e * EDGE_DIM + lane] : 0.0f;
        float ev[N_HEADS];
        #pragma unroll
        for (int h = 0; h < N_HEADS; h++) ev[h] = 0.0f;
        for (int j = 0; j < EDGE_DIM; j++) {
            float aj = __shfl(ear, j, 32);
            #pragma unroll
            for (int h = 0; h < N_HEADS; h++)
                ev[h] = fmaf(aj, sWe[j * HD + h * 32 + lane], ev[h]);
        }
        const float* vr = V + (long long)src * HD;
        #pragma unroll
        for (int h = 0; h < N_HEADS; h++) {
            float lg = logits[(long long)e * N_HEADS + h];          // broadcast load
            float m  = funkey2(mkey[(long long)dst * N_HEADS + h]); // broadcast load
            float a  = __expf(lg - m);
            if (lane == 0) atomicAdd(&denom[(long long)dst * N_HEADS + h], a);
            float vj = vr[h * 32 + lane] + ev[h];
            atomicAdd(&outacc[(long long)dst * HD + h * 32 + lane], a * vj);
        }
    }
}

// ---------------- finalize: normalize + skip + relu ----------------
__global__ void finalize_kernel(const float* __restrict__ outacc,
                                const float* __restrict__ denom,
                                const float* __restrict__ skip,
                                float* __restrict__ hout, int nrows) {
    long long idx = (long long)blockIdx.x * blockDim.x + threadIdx.x;
    long long n = (long long)nrows * HD;
    if (idx >= n) return;
    int node = (int)(idx >> 8);          // /256
    int ch   = (int)(idx & 255);
    int h    = ch >> 5;
    float dn = denom[(long long)node * N_HEADS + h];
    float v  = (dn > 0.0f) ? (outacc[idx] / dn) : 0.0f;
    v += skip[idx];
    hout[idx] = fmaxf(v, 0.0f);
}

// ---------------- global mean pool (accumulate) ----------------
__global__ void pool_kernel(const float* __restrict__ h,
                            const int* __restrict__ batch,
                            float* __restrict__ pooled,
                            float* __restrict__ cnt, int nrows) {
    long long idx = (long long)blockIdx.x * blockDim.x + threadIdx.x;
    long long n = (long long)nrows * HD;
    if (idx >= n) return;
    int node = (int)(idx >> 8);
    int ch   = (int)(idx & 255);
    int g    = batch[node];
    atomicAdd(&pooled[(long long)g * HD + ch], h[idx]);
    if (ch == 0) atomicAdd(&cnt[g], 1.0f);
}

// ---------------- final FC: out[g,c] = (pooled[g]/cnt) @ Wfc + bfc ----------------
__global__ void fc_kernel(const float* __restrict__ pooled,
                          const float* __restrict__ cnt,
                          const float* __restrict__ Wfc,
                          const float* __restrict__ bfc,
                          float* __restrict__ out) {
    int idx = blockIdx.x * blockDim.x + threadIdx.x;
    if (idx >= N_GRAPHS * N_CLASSES) return;
    int g = idx >> 5;          // /32
    int c = idx & 31;
    float inv = 1.0f / fmaxf(cnt[g], 1.0f);
    float s = bfc[c];
    for (int k = 0; k < HD; k++)
        s = fmaf(pooled[(long long)g * HD + k] * inv, Wfc[k * N_CLASSES + c], s);
    out[idx] = s;
}

// ---------------------------------------------------------------------------
extern "C" void kernel_launch(void* const* d_in, const int* in_sizes, int n_in,
                              void* d_out, int out_size, void* d_ws, size_t ws_size,
                              hipStream_t stream) {
    (void)in_sizes; (void)n_in; (void)out_size; (void)ws_size;

    // ---- inputs (top-level dict order; params flattened in insertion order)
    const float* x   = (const float*)d_in[0];             // [N,128]
    const int*   ei  = (const int*)d_in[1];               // [2,E]
    const float* ea  = (const float*)d_in[2];             // [E,26]
    const int*   bat = (const int*)d_in[3];               // [N]
    // layer L params at d_in[4 + 9*L + {0..8}] = Wq,bq,Wk,bk,Wv,bv,We,Ws,bs
    const float* Wfc = (const float*)d_in[4 + 9 * 3];     // [256,32]
    const float* bfc = (const float*)d_in[4 + 9 * 3 + 1]; // [32]

    // ---- workspace carve (floats)
    const long long NHD = (long long)N_NODES * HD;        // 5,120,000
    float* ws = (float*)d_ws;
    float*    Q      = ws;               ws += NHD;
    float*    Kf     = ws;               ws += NHD;
    float*    Vf     = ws;               ws += NHD;
    float*    Sf     = ws;               ws += NHD;
    float*    OutAcc = ws;               ws += NHD;       // contiguous with mkey,denom
    unsigned* mkey   = (unsigned*)ws;    ws += (long long)N_NODES * N_HEADS;
    float*    denom  = ws;               ws += (long long)N_NODES * N_HEADS;
    float*    logits = ws;               ws += (long long)N_EDGES * N_HEADS;
    float*    h1     = ws;               ws += NHD;
    float*    h2     = ws;               ws += NHD;
    float*    pooled = ws;               ws += (long long)N_GRAPHS * HD;   // contiguous with cnt
    float*    cnt    = ws;               ws += N_GRAPHS;

    const float* hin = x;
    int fin = F_IN;
    float* houts[3] = { h1, h2, h1 };

    const int gemm_bx = (N_NODES + 63) / 64;   // 313
    const int edge_gx = (N_EDGES + 63) / 64;   // 5000
    const int fin_gx  = (int)((NHD + 255) / 256);

    for (int L = 0; L < 3; L++) {
        const float* Wq = (const float*)d_in[4 + 9 * L + 0];
        const float* bq = (const float*)d_in[4 + 9 * L + 1];
        const float* Wk = (const float*)d_in[4 + 9 * L + 2];
        const float* bk = (const float*)d_in[4 + 9 * L + 3];
        const float* Wv = (const float*)d_in[4 + 9 * L + 4];
        const float* bv = (const float*)d_in[4 + 9 * L + 5];
        const float* We = (const float*)d_in[4 + 9 * L + 6];
        const float* Ws = (const float*)d_in[4 + 9 * L + 7];
        const float* bs = (const float*)d_in[4 + 9 * L + 8];

        // zero OutAcc + mkey + denom in one shot (contiguous)
        long long zn = NHD + 2LL * N_NODES * N_HEADS;
        zero_f32_kernel<<<1024, 256, 0, stream>>>(OutAcc, zn);

        gemm4_kernel<<<dim3(gemm_bx, 16, 4), 128, 0, stream>>>(
            hin, fin, N_NODES,
            Wq, bq, Q,  Wk, bk, Kf,  Wv, bv, Vf,  Ws, bs, Sf);

        edge_logits_kernel<<<edge_gx, 256, 0, stream>>>(
            ei, ea, Q, Kf, We, logits, mkey, N_EDGES);

        edge_accum_kernel<<<edge_gx, 256, 0, stream>>>(
            ei, ea, Vf, We, logits, mkey, denom, OutAcc, N_EDGES);

        finalize_kernel<<<fin_gx, 256, 0, stream>>>(
            OutAcc, denom, Sf, houts[L], N_NODES);

        hin = houts[L];
        fin = HD;
    }

    // pooling + FC
    zero_f32_kernel<<<64, 256, 0, stream>>>(pooled, (long long)N_GRAPHS * HD + N_GRAPHS);
    pool_kernel<<<fin_gx, 256, 0, stream>>>(houts[2], bat, pooled, cnt, N_NODES);
    fc_kernel<<<(N_GRAPHS * N_CLASSES + 255) / 256, 256, 0, stream>>>(
        pooled, cnt, Wfc, bfc, (float*)d_out);
}